// GCN_12773232738508
// MI455X (gfx1250) — compile-verified
//
#include <hip/hip_runtime.h>

typedef float v2f __attribute__((ext_vector_type(2)));
typedef float v8f __attribute__((ext_vector_type(8)));

#define KF 256           // feature dim entering every layer's scatter + GEMM K
#define NS0 1081344
#define ND0 67584
#define ND1 6144
#define ND2 1024

// ---------------- utility kernels ----------------

__global__ void GCN_zero_u32(unsigned* __restrict__ p, unsigned n) {
  unsigned i = blockIdx.x * blockDim.x + threadIdx.x;
  if (i < n) p[i] = 0u;
}

__global__ void GCN_count(const int* __restrict__ idx, unsigned n,
                          unsigned* __restrict__ cnt) {
  unsigned i = blockIdx.x * blockDim.x + threadIdx.x;
  if (i < n) atomicAdd(&cnt[idx[i]], 1u);
}

// in-place: uint degree count -> float rsqrt(max(deg,1))
__global__ void GCN_rsqrt(unsigned* __restrict__ buf, unsigned n) {
  unsigned i = blockIdx.x * blockDim.x + threadIdx.x;
  if (i < n) {
    unsigned c = buf[i];
    float deg = (c > 0u) ? (float)c : 1.0f;
    ((float*)buf)[i] = rsqrtf(deg);
  }
}

// ---------------- edge scatter: agg[dst] += x[src] * dnorm_src[src] ----------------
// 64 threads per edge, 4 contiguous feats per thread (float4 load, native f32
// atomics into an L2-resident accumulation buffer).

__global__ void GCN_scatter(const float* __restrict__ x, const int* __restrict__ src,
                            const int* __restrict__ dst, const float* __restrict__ dnorm,
                            float* __restrict__ agg, unsigned n_edges) {
  unsigned gid = blockIdx.x * blockDim.x + threadIdx.x;
  unsigned e = gid >> 6;
  if (e >= n_edges) return;
  unsigned c = (gid & 63u) << 2;
  int s = src[e];
  int d = dst[e];
  float w = dnorm[s];
  const float4 v = *(const float4*)(x + (size_t)s * KF + c);
  float* a = agg + (size_t)d * KF + c;
  unsafeAtomicAdd(a + 0, v.x * w);
  unsafeAtomicAdd(a + 1, v.y * w);
  unsafeAtomicAdd(a + 2, v.z * w);
  unsafeAtomicAdd(a + 3, v.w * w);
}

// ---------------- WMMA GEMM with fused epilogue ----------------
// out[r, c] = act( (A[r,:] @ W[:,c]) * dnorm_row[r] + bias[c] )
// One wave32 computes a 16 x (16*NT) output strip: NT accumulators (NT*8 VGPRs),
// each A v2f load feeds NT chained V_WMMA_F32_16X16X4_F32 (K=4 -> 64 steps).
// A: [M x 256] row-major, W: [256 x N] row-major, M multiple of 16.
// MASK=false requires all NT*16 columns of the group to be in-bounds.

template <int NT, bool MASK>
__global__ void GCN_gemm(const float* __restrict__ A, const float* __restrict__ W,
                         const float* __restrict__ bias, const float* __restrict__ dnorm,
                         float* __restrict__ out, int ngroups_n, int total_groups,
                         int N, int relu) {
  int wave = ((int)blockIdx.x * (int)blockDim.x + (int)threadIdx.x) >> 5;
  if (wave >= total_groups) return;          // wave-uniform: EXEC stays all-1s
  int lane = (int)threadIdx.x & 31;
  int m    = lane & 15;
  int half = lane >> 4;                      // 0: K=0,1 / rows 0-7 ; 1: K=2,3 / rows 8-15
  int gm = wave / ngroups_n;
  int gn = wave % ngroups_n;

  // A operand: lane m (both halves) holds row gm*16+m, this half's K pair
  const float* ap = A + (size_t)(gm * 16 + m) * KF + half * 2;

  int col[NT];
  float cmsk[NT];
  const float* wp[NT];
#pragma unroll
  for (int t = 0; t < NT; ++t) {
    col[t] = (gn * NT + t) * 16 + m;
    if (MASK) {
      bool ok = col[t] < N;
      cmsk[t] = ok ? 1.0f : 0.0f;
      wp[t] = W + (size_t)(half * 2) * N + (ok ? col[t] : (N - 1));
    } else {
      cmsk[t] = 1.0f;
      wp[t] = W + (size_t)(half * 2) * N + col[t];
    }
  }

  v8f acc[NT];
#pragma unroll
  for (int t = 0; t < NT; ++t) acc[t] = (v8f){};

#pragma unroll 4
  for (int k = 0; k < KF; k += 4) {
    v2f a;
    a.x = ap[k];
    a.y = ap[k + 1];
#pragma unroll
    for (int t = 0; t < NT; ++t) {
      v2f b;
      if (MASK) {
        b.x = wp[t][(size_t)k * N] * cmsk[t];
        b.y = wp[t][(size_t)(k + 1) * N] * cmsk[t];
      } else {
        b.x = wp[t][(size_t)k * N];
        b.y = wp[t][(size_t)(k + 1) * N];
      }
      acc[t] = __builtin_amdgcn_wmma_f32_16x16x4_f32(
          /*neg_a=*/false, a, /*neg_b=*/false, b,
          /*c_mod=*/(short)0, acc[t], /*reuse_a=*/false, /*reuse_b=*/false);
    }
  }

#pragma unroll
  for (int t = 0; t < NT; ++t) {
    float bcol = MASK ? (bias[col[t] < N ? col[t] : (N - 1)] * cmsk[t]) : bias[col[t]];
#pragma unroll
    for (int r = 0; r < 8; ++r) {
      int row = gm * 16 + half * 8 + r;      // C/D layout: VGPR r -> M = r + half*8
      float v = acc[t][r] * dnorm[row] + bcol;
      if (relu) v = fmaxf(v, 0.0f);
      if (!MASK || col[t] < N) out[(size_t)row * N + col[t]] = v;
    }
  }
}

// ---------------- host orchestration ----------------

extern "C" void kernel_launch(void* const* d_in, const int* in_sizes, int n_in,
                              void* d_out, int out_size, void* d_ws, size_t ws_size,
                              hipStream_t stream) {
  const float* x   = (const float*)d_in[0];
  const int* src0  = (const int*)d_in[1];
  const int* dst0  = (const int*)d_in[2];
  const int* src1  = (const int*)d_in[3];
  const int* dst1  = (const int*)d_in[4];
  const int* src2  = (const int*)d_in[5];
  const int* dst2  = (const int*)d_in[6];
  const float* W0  = (const float*)d_in[7];
  const float* b0  = (const float*)d_in[8];
  const float* W1  = (const float*)d_in[9];
  const float* b1  = (const float*)d_in[10];
  const float* W2  = (const float*)d_in[11];
  const float* b2  = (const float*)d_in[12];

  const unsigned e0 = (unsigned)in_sizes[1];   // 1,013,760
  const unsigned e1 = (unsigned)in_sizes[3];   //    61,440
  const unsigned e2 = (unsigned)in_sizes[5];   //     5,120

  // workspace: two 69MB ping-pong feature buffers + degree/norm arrays (~137MB)
  char* ws = (char*)d_ws;
  const size_t BUF = (size_t)ND0 * KF * sizeof(float);   // 69,206,016 B
  float* bufA = (float*)ws;                              // agg buffers (reused per layer)
  float* bufB = (float*)(ws + BUF);                      // hidden activations (reused)
  unsigned* cnt = (unsigned*)(ws + 2 * BUF);
  unsigned* cs0 = cnt;                // NS0
  unsigned* cd0 = cs0 + NS0;          // ND0
  unsigned* cs1 = cd0 + ND0;          // ND0
  unsigned* cd1 = cs1 + ND0;          // ND1
  unsigned* cs2 = cd1 + ND1;          // ND1
  unsigned* cd2 = cs2 + ND1;          // ND2
  const unsigned total_norm = NS0 + ND0 + ND0 + ND1 + ND1 + ND2;  // 1,229,824

  const int B = 256;
  auto blocks = [](unsigned n, int b) { return (unsigned)((n + b - 1) / b); };

  // 1) degrees -> rsqrt norms
  GCN_zero_u32<<<blocks(total_norm, B), B, 0, stream>>>(cnt, total_norm);
  GCN_count<<<blocks(e0, B), B, 0, stream>>>(src0, e0, cs0);
  GCN_count<<<blocks(e0, B), B, 0, stream>>>(dst0, e0, cd0);
  GCN_count<<<blocks(e1, B), B, 0, stream>>>(src1, e1, cs1);
  GCN_count<<<blocks(e1, B), B, 0, stream>>>(dst1, e1, cd1);
  GCN_count<<<blocks(e2, B), B, 0, stream>>>(src2, e2, cs2);
  GCN_count<<<blocks(e2, B), B, 0, stream>>>(dst2, e2, cd2);
  GCN_rsqrt<<<blocks(total_norm, B), B, 0, stream>>>(cnt, total_norm);

  // 2) layer 0: aggregate-first, then [67584x256]@[256x256] WMMA GEMM (+relu)
  GCN_zero_u32<<<blocks((unsigned)ND0 * KF, B), B, 0, stream>>>((unsigned*)bufA, (unsigned)ND0 * KF);
  GCN_scatter<<<blocks(e0 * 64u, B), B, 0, stream>>>(x, src0, dst0, (const float*)cs0, bufA, e0);
  {
    int ngn = 256 / (16 * 8);                 // 2 column groups of 128
    int groups = (ND0 / 16) * ngn;            // 8,448 waves
    GCN_gemm<8, false><<<blocks((unsigned)groups * 32u, B), B, 0, stream>>>(
        bufA, W0, b0, (const float*)cd0, bufB, ngn, groups, 256, 1);
  }

  // 3) layer 1: [6144x256]@[256x256] (+relu)
  GCN_zero_u32<<<blocks((unsigned)ND1 * KF, B), B, 0, stream>>>((unsigned*)bufA, (unsigned)ND1 * KF);
  GCN_scatter<<<blocks(e1 * 64u, B), B, 0, stream>>>(bufB, src1, dst1, (const float*)cs1, bufA, e1);
  {
    int ngn = 256 / (16 * 8);
    int groups = (ND1 / 16) * ngn;            // 768 waves
    GCN_gemm<8, false><<<blocks((unsigned)groups * 32u, B), B, 0, stream>>>(
        bufA, W1, b1, (const float*)cd1, bufB, ngn, groups, 256, 1);
  }

  // 4) layer 2: [1024x256]@[256x47], no activation, straight to d_out
  GCN_zero_u32<<<blocks((unsigned)ND2 * KF, B), B, 0, stream>>>((unsigned*)bufA, (unsigned)ND2 * KF);
  GCN_scatter<<<blocks(e2 * 64u, B), B, 0, stream>>>(bufB, src2, dst2, (const float*)cs2, bufA, e2);
  {
    int groups = ND2 / 16;                    // 64 waves, NT=3 covers N=47
    GCN_gemm<3, true><<<blocks((unsigned)groups * 32u, B), B, 0, stream>>>(
        bufA, W2, b2, (const float*)cd2, (float*)d_out, 1, groups, 47, 0);
  }
}